// LQR_Solver_21947282882796
// MI455X (gfx1250) — compile-verified
//
#include <hip/hip_runtime.h>

typedef float v2f __attribute__((ext_vector_type(2)));
typedef float v8f __attribute__((ext_vector_type(8)));

#define NS 32
#define NI 16
#define NA 48
#define HT 64          // horizon T

#define LDAB 49
#define LDP  33
#define LDW  49
#define LDQ  49

__device__ __forceinline__ v8f wmma4(v2f a, v2f b, v8f c) {
  // D = A(16x4,f32) * B(4x16,f32) + C(16x16,f32)
  return __builtin_amdgcn_wmma_f32_16x16x4_f32(false, a, false, b, (short)0, c,
                                               false, false);
}

__global__ __launch_bounds__(128) void lqr_riccati_kernel(
    const float* __restrict__ Ag, const float* __restrict__ Bg,
    const float* __restrict__ Cg, const float* __restrict__ x0g,
    float* __restrict__ out, float* __restrict__ Kall, float* __restrict__ Pall)
{
  __shared__ float sAB[32 * LDAB];   // [A | B]  (32 x 48)
  __shared__ float sP [32 * LDP];    // P_{t+1}  (32 x 32)
  __shared__ float sW [32 * LDW];    // P * AB   (32 x 48)
  __shared__ float sQ [48 * LDQ];    // Q_t      (48 x 48)
  __shared__ float sx[64];           // [x ; u]

  const int tid  = threadIdx.x;
  const int wave = tid >> 5;
  const int lane = tid & 31;
  const int m  = lane & 15;          // tile row (A frag) / tile col (B,C frags)
  const int kh = lane >> 4;          // which K-pair this half-wave owns
  const int rr = tid >> 3;           // GJ: fixed row 0..15 (8 threads per row)
  const int c0 = (tid & 7) * 6;      // GJ: fixed 6-column strip, covers 0..47

  // ---- stage AB = [A | B], zero P_T ----
  for (int e = tid; e < 32 * 48; e += 128) {
    int i = e / 48, j = e % 48;
    sAB[i * LDAB + j] = (j < 32) ? Ag[i * 32 + j] : Bg[i * 16 + (j - 32)];
  }
  for (int e = tid; e < 32 * 32; e += 128) sP[(e / 32) * LDP + (e % 32)] = 0.f;
  __syncthreads();

  // ================= backward Riccati sweep =================
  for (int t = HT - 1; t >= 0; --t) {
    // C_t -> sQ (serves as the WMMA accumulator init for Q); coalesced rows
    const float* Ct = Cg + (size_t)t * (NA * NA);
    for (int r = wave; r < NA; r += 4) {
      sQ[r * LDQ + lane] = Ct[r * NA + lane];
      if (lane < 16) sQ[r * LDQ + 32 + lane] = Ct[r * NA + 32 + lane];
    }
    if (t > 0)  // hide latency of next iteration's C block
      __builtin_prefetch(Cg + (size_t)(t - 1) * (NA * NA) + tid * 18, 0, 1);

    // W = P * AB  (32x48, K=32): 6 tiles across 4 waves
    for (int tt = wave; tt < 6; tt += 4) {
      const int r0 = (tt / 3) * 16, cc = (tt % 3) * 16;
      v8f acc; for (int r = 0; r < 8; ++r) acc[r] = 0.f;
      for (int k0 = 0; k0 < 32; k0 += 4) {
        v2f a, b;
        a[0] = sP[(r0 + m) * LDP + k0 + 2 * kh + 0];
        a[1] = sP[(r0 + m) * LDP + k0 + 2 * kh + 1];
        b[0] = sAB[(k0 + 2 * kh + 0) * LDAB + cc + m];
        b[1] = sAB[(k0 + 2 * kh + 1) * LDAB + cc + m];
        acc = wmma4(a, b, acc);
      }
      for (int r = 0; r < 8; ++r)
        sW[(r0 + r + 8 * kh) * LDW + cc + m] = acc[r];
    }
    __syncthreads();

    // Q = C_t + AB^T * W  (48x48, K=32): 9 tiles
    for (int tt = wave; tt < 9; tt += 4) {
      const int r0 = (tt / 3) * 16, cc = (tt % 3) * 16;
      v8f acc;
      for (int r = 0; r < 8; ++r) acc[r] = sQ[(r0 + r + 8 * kh) * LDQ + cc + m];
      for (int k0 = 0; k0 < 32; k0 += 4) {
        v2f a, b;
        a[0] = sAB[(k0 + 2 * kh + 0) * LDAB + r0 + m];  // AB^T(m,k) = AB(k,m)
        a[1] = sAB[(k0 + 2 * kh + 1) * LDAB + r0 + m];
        b[0] = sW[(k0 + 2 * kh + 0) * LDW + cc + m];
        b[1] = sW[(k0 + 2 * kh + 1) * LDW + cc + m];
        acc = wmma4(a, b, acc);
      }
      for (int r = 0; r < 8; ++r)
        sQ[(r0 + r + 8 * kh) * LDQ + cc + m] = acc[r];
    }

    // ---- Jordan elimination on rows 32..47 (Quu SPD, no pivoting) ----
    // Unscaled elimination: pivot row is never written while being read.
    // Afterwards Quu is diagonal; divide by the diagonal to leave
    // K_t = Quu^{-1} Qux in sQ[32..47][0..31].
    for (int j = 0; j < 16; ++j) {
      __syncthreads();                 // previous writes visible (1st iter: Q-phase)
      const float rp = 1.0f / sQ[(32 + j) * LDQ + 32 + j];   // LDS broadcast
      const float f  = sQ[(32 + rr) * LDQ + 32 + j] * rp;
      __syncthreads();                 // all factor reads done before any write
      if (rr != j) {
#pragma unroll
        for (int i = 0; i < 6; ++i)
          sQ[(32 + rr) * LDQ + c0 + i] -= f * sQ[(32 + j) * LDQ + c0 + i];
      }
    }
    __syncthreads();
    {
      const float dinv = 1.0f / sQ[(32 + rr) * LDQ + 32 + rr];
#pragma unroll
      for (int i = 0; i < 6; ++i)      // 8-thread row group sits in one wave: lockstep-safe
        sQ[(32 + rr) * LDQ + c0 + i] *= dinv;
    }
    __syncthreads();

    // P_new = Qxx - Qxu * K_t  (32x32, K=16): 4 tiles, one per wave
    {
      const int r0 = (wave >> 1) * 16, cc = (wave & 1) * 16;
      v8f acc;
      for (int r = 0; r < 8; ++r) acc[r] = sQ[(r0 + r + 8 * kh) * LDQ + cc + m];
      for (int k0 = 0; k0 < 16; k0 += 4) {
        v2f a, b;
        a[0] = -sQ[(r0 + m) * LDQ + 32 + k0 + 2 * kh + 0];     // -Qxu
        a[1] = -sQ[(r0 + m) * LDQ + 32 + k0 + 2 * kh + 1];
        b[0] = sQ[(32 + k0 + 2 * kh + 0) * LDQ + cc + m];      // K_t
        b[1] = sQ[(32 + k0 + 2 * kh + 1) * LDQ + cc + m];
        acc = wmma4(a, b, acc);
      }
      for (int r = 0; r < 8; ++r) {
        const int r2 = r0 + r + 8 * kh, c2 = cc + m;
        sP[r2 * LDP + c2] = acc[r];
        Pall[t * 1024 + r2 * 32 + c2] = acc[r];
      }
    }
    // save K_t for the forward rollout (shift/mask indexing only)
    for (int e = tid; e < 16 * 32; e += 128)
      Kall[t * 512 + e] = sQ[(32 + (e >> 5)) * LDQ + (e & 31)];
    __syncthreads();
  }

  __threadfence();
  __syncthreads();

  // ================= forward rollout + multipliers =================
  if (tid < 32) sx[tid] = x0g[tid];
  __syncthreads();

  for (int t = 0; t < HT; ++t) {
    const float* Kt = Kall + t * 512;
    const float* Pt = Pall + t * 1024;
    if (tid < 16) {                                  // u_t = -K_t x_t
      float s = 0.f;
      for (int k = 0; k < 32; ++k) s += Kt[tid * 32 + k] * sx[k];
      sx[32 + tid] = -s;
    }
    __syncthreads();
    if (tid < 48) out[t * 48 + tid] = sx[tid];       // tau_t = [x_t; u_t]
    if (tid < 32) {                                  // mu_t = P_t x_t (t=0: negated)
      float mu = 0.f;
      for (int k = 0; k < 32; ++k) mu += Pt[tid * 32 + k] * sx[k];
      if (t == 0) mu = -mu;
      out[NA * HT + t * 32 + tid] = mu;
    }
    float xn = 0.f;
    if (tid < 32) {                                  // x_{t+1} = [A B] [x;u]
      for (int k = 0; k < 48; ++k) xn += sAB[tid * LDAB + k] * sx[k];
    }
    __syncthreads();
    if (tid < 32 && t < HT - 1) sx[tid] = xn;
    __syncthreads();
  }
}

extern "C" void kernel_launch(void* const* d_in, const int* in_sizes, int n_in,
                              void* d_out, int out_size, void* d_ws, size_t ws_size,
                              hipStream_t stream) {
  (void)in_sizes; (void)n_in; (void)out_size; (void)ws_size;
  const float* A  = (const float*)d_in[0];
  const float* B  = (const float*)d_in[1];
  const float* C  = (const float*)d_in[2];
  // d_in[3] = c (unused by the reference RHS), d_in[4] = T (fixed 64)
  const float* x0 = (const float*)d_in[5];
  float* out  = (float*)d_out;
  float* Kall = (float*)d_ws;                  // 64 * 16*32 floats = 128 KB
  float* Pall = Kall + HT * NI * NS;           // 64 * 32*32 floats = 256 KB
  lqr_riccati_kernel<<<dim3(1), dim3(128), 0, stream>>>(A, B, C, x0, out, Kall, Pall);
}